// GCN_16501264351680
// MI455X (gfx1250) — compile-verified
//
#include <hip/hip_runtime.h>
#include <hip/hip_bf16.h>
#include <math.h>

typedef float v2f __attribute__((ext_vector_type(2)));
typedef float v8f __attribute__((ext_vector_type(8)));

#define N_NODES 100000
#define N_EDGES 3200000
#define NT1 (N_NODES / 16)   // 6250 exact tiles of 16 nodes

// ---------------- degree / norm ----------------
__global__ void k_init_deg(float* __restrict__ deg) {
  int i = blockIdx.x * blockDim.x + threadIdx.x;
  if (i < N_NODES) deg[i] = 1.0f;  // self-loop contributes 1 to degree
}

__global__ void k_deg_edges(const long long* __restrict__ ei, float* __restrict__ deg) {
  int e = blockIdx.x * blockDim.x + threadIdx.x;
  if (e < N_EDGES) {
    long long d = ei[(long long)N_EDGES + e];  // dst row
    atomicAdd(&deg[d], 1.0f);
  }
}

__global__ void k_dinv(float* __restrict__ deg) {
  int i = blockIdx.x * blockDim.x + threadIdx.x;
  if (i < N_NODES) {
    float d = deg[i];
    deg[i] = (d > 0.0f) ? rsqrtf(d) : 0.0f;  // in place: deg -> dinv
  }
}

// ---------------- layer-1 GEMM: h1 = x @ W1 via V_WMMA_F32_16X16X4_F32 ----------------
// Each wave computes one 16-node x 16-feature tile; K=256 -> 64 wmma steps.
__global__ void k_gemm1_wmma(const float* __restrict__ x,
                             const float* __restrict__ W1,
                             float* __restrict__ h1) {
  const int lane = threadIdx.x & 31;
  const int wave = threadIdx.x >> 5;
  const int tile = blockIdx.x * (blockDim.x >> 5) + wave;
  if (tile >= NT1) return;                 // uniform per-wave: EXEC stays all-ones
  const int n  = lane & 15;                // A: row M  |  B/D: column N
  const int kb = (lane >> 4) << 1;         // K sub-offset: 0 or 2
  const float* __restrict__ xrow = x + (size_t)(tile * 16 + n) * 256;

  v8f acc = {0.f, 0.f, 0.f, 0.f, 0.f, 0.f, 0.f, 0.f};
#pragma unroll 8
  for (int k0 = 0; k0 < 256; k0 += 4) {
    v2f a, b;
    a.x = xrow[k0 + kb];
    a.y = xrow[k0 + kb + 1];
    b.x = W1[(k0 + kb)     * 16 + n];
    b.y = W1[(k0 + kb + 1) * 16 + n];
    acc = __builtin_amdgcn_wmma_f32_16x16x4_f32(false, a, false, b,
                                                (short)0, acc, false, false);
  }
  const int hi = lane >> 4;                // D: VGPR r holds M=r (lo half) / r+8 (hi half)
#pragma unroll
  for (int r = 0; r < 8; ++r) {
    int m = tile * 16 + r + 8 * hi;
    h1[(size_t)m * 16 + n] = acc[r];
  }
}

// ---------------- self-loop init: agg[i] = dinv[i]^2 * h[i] (writes whole buffer) ----------------
__global__ void k_self1(const float* __restrict__ dinv, const float* __restrict__ h1,
                        float* __restrict__ agg1) {
  int t = blockIdx.x * blockDim.x + threadIdx.x;   // N_NODES * 4 threads (float4 each)
  if (t >= N_NODES * 4) return;
  int i = t >> 2, f = (t & 3) * 4;
  float s = dinv[i]; s = s * s;
  float4 v = *(const float4*)(h1 + (size_t)i * 16 + f);
  float4 o = make_float4(v.x * s, v.y * s, v.z * s, v.w * s);
  *(float4*)(agg1 + (size_t)i * 16 + f) = o;
}

// ---------------- edge scatter layer 1: 4 threads per edge, 4 feats each ----------------
__global__ void k_scatter1(const long long* __restrict__ ei, const float* __restrict__ dinv,
                           const float* __restrict__ h1, float* __restrict__ agg1) {
  int t = blockIdx.x * blockDim.x + threadIdx.x;   // 4 * N_EDGES = 12.8M
  if (t >= 4 * N_EDGES) return;
  int e = t >> 2, f = (t & 3) * 4;
  long long s = ei[e];
  long long d = ei[(long long)N_EDGES + e];
  float norm = dinv[s] * dinv[d];
  float4 v = *(const float4*)(h1 + (size_t)s * 16 + f);
  float* o = agg1 + (size_t)d * 16 + f;
  atomicAdd(o + 0, v.x * norm);
  atomicAdd(o + 1, v.y * norm);
  atomicAdd(o + 2, v.z * norm);
  atomicAdd(o + 3, v.w * norm);
}

// ---------------- h = relu(agg1 + b1), in place ----------------
__global__ void k_relu_bias(float* __restrict__ h, const float* __restrict__ b1) {
  int t = blockIdx.x * blockDim.x + threadIdx.x;
  if (t >= N_NODES * 4) return;
  int i = t >> 2, f = (t & 3) * 4;
  float4 v = *(float4*)(h + (size_t)i * 16 + f);
  v.x = fmaxf(v.x + b1[f + 0], 0.f);
  v.y = fmaxf(v.y + b1[f + 1], 0.f);
  v.z = fmaxf(v.z + b1[f + 2], 0.f);
  v.w = fmaxf(v.w + b1[f + 3], 0.f);
  *(float4*)(h + (size_t)i * 16 + f) = v;
}

// ---------------- layer-2 GEMM: h2 = h @ W2 (16x8, tiny -> VALU) ----------------
__global__ void k_gemm2(const float* __restrict__ h, const float* __restrict__ W2,
                        float* __restrict__ h2) {
  int i = blockIdx.x * blockDim.x + threadIdx.x;
  if (i >= N_NODES) return;
  float hr[16];
  const float4* hp = (const float4*)(h + (size_t)i * 16);
#pragma unroll
  for (int q = 0; q < 4; ++q) {
    float4 v = hp[q];
    hr[q * 4 + 0] = v.x; hr[q * 4 + 1] = v.y; hr[q * 4 + 2] = v.z; hr[q * 4 + 3] = v.w;
  }
#pragma unroll
  for (int j = 0; j < 8; ++j) {
    float acc = 0.f;
#pragma unroll
    for (int k = 0; k < 16; ++k) acc = fmaf(hr[k], W2[k * 8 + j], acc);
    h2[(size_t)i * 8 + j] = acc;
  }
}

__global__ void k_self2(const float* __restrict__ dinv, const float* __restrict__ h2,
                        float* __restrict__ agg2) {
  int t = blockIdx.x * blockDim.x + threadIdx.x;   // N_NODES * 2 (float4 each)
  if (t >= N_NODES * 2) return;
  int i = t >> 1, f = (t & 1) * 4;
  float s = dinv[i]; s = s * s;
  float4 v = *(const float4*)(h2 + (size_t)i * 8 + f);
  float4 o = make_float4(v.x * s, v.y * s, v.z * s, v.w * s);
  *(float4*)(agg2 + (size_t)i * 8 + f) = o;
}

__global__ void k_scatter2(const long long* __restrict__ ei, const float* __restrict__ dinv,
                           const float* __restrict__ h2, float* __restrict__ agg2) {
  int t = blockIdx.x * blockDim.x + threadIdx.x;   // 2 * N_EDGES
  if (t >= 2 * N_EDGES) return;
  int e = t >> 1, f = (t & 1) * 4;
  long long s = ei[e];
  long long d = ei[(long long)N_EDGES + e];
  float norm = dinv[s] * dinv[d];
  float4 v = *(const float4*)(h2 + (size_t)s * 8 + f);
  float* o = agg2 + (size_t)d * 8 + f;
  atomicAdd(o + 0, v.x * norm);
  atomicAdd(o + 1, v.y * norm);
  atomicAdd(o + 2, v.z * norm);
  atomicAdd(o + 3, v.w * norm);
}

// ---------------- out = log_softmax(agg2 + b2), in place on d_out ----------------
__global__ void k_logsoftmax(float* __restrict__ out, const float* __restrict__ b2) {
  int i = blockIdx.x * blockDim.x + threadIdx.x;
  if (i >= N_NODES) return;
  float* row = out + (size_t)i * 8;
  float v[8], m = -1e30f;
#pragma unroll
  for (int j = 0; j < 8; ++j) { v[j] = row[j] + b2[j]; m = fmaxf(m, v[j]); }
  float s = 0.f;
#pragma unroll
  for (int j = 0; j < 8; ++j) s += expf(v[j] - m);
  float l = logf(s);
#pragma unroll
  for (int j = 0; j < 8; ++j) row[j] = v[j] - m - l;
}

extern "C" void kernel_launch(void* const* d_in, const int* in_sizes, int n_in,
                              void* d_out, int out_size, void* d_ws, size_t ws_size,
                              hipStream_t stream) {
  const float*      x  = (const float*)d_in[0];
  const long long*  ei = (const long long*)d_in[1];   // int64 [2, E] row-major
  const float*      W1 = (const float*)d_in[2];       // [256,16]
  const float*      b1 = (const float*)d_in[3];
  const float*      W2 = (const float*)d_in[4];       // [16,8]
  const float*      b2 = (const float*)d_in[5];
  float*            out = (float*)d_out;              // [N,8]; used as agg2 accumulator

  char* ws = (char*)d_ws;
  float* dinv = (float*)(ws);                 // N floats (deg -> dinv in place)
  float* h1   = (float*)(ws + (1u  << 20));   // N*16 floats (6.4 MB)
  float* agg1 = (float*)(ws + (8u  << 20));   // N*16 floats; becomes h after relu
  float* h2   = (float*)(ws + (15u << 20));   // N*8 floats (3.2 MB)

  const int B = 256;
  k_init_deg   <<<(N_NODES + B - 1) / B, B, 0, stream>>>(dinv);
  k_deg_edges  <<<(N_EDGES + B - 1) / B, B, 0, stream>>>(ei, dinv);
  k_dinv       <<<(N_NODES + B - 1) / B, B, 0, stream>>>(dinv);

  // layer 1
  k_gemm1_wmma <<<(NT1 + 7) / 8, B, 0, stream>>>(x, W1, h1);       // 8 waves/block
  k_self1      <<<(N_NODES * 4 + B - 1) / B, B, 0, stream>>>(dinv, h1, agg1);
  k_scatter1   <<<(N_EDGES * 4 + B - 1) / B, B, 0, stream>>>(ei, dinv, h1, agg1);
  k_relu_bias  <<<(N_NODES * 4 + B - 1) / B, B, 0, stream>>>(agg1, b1);

  // layer 2
  k_gemm2      <<<(N_NODES + B - 1) / B, B, 0, stream>>>(agg1, W2, h2);
  k_self2      <<<(N_NODES * 2 + B - 1) / B, B, 0, stream>>>(dinv, h2, out);
  k_scatter2   <<<(N_EDGES * 2 + B - 1) / B, B, 0, stream>>>(ei, dinv, h2, out);
  k_logsoftmax <<<(N_NODES + B - 1) / B, B, 0, stream>>>(out, b2);
}